// SmoothAP_20023137534430
// MI455X (gfx1250) — compile-verified
//
#include <hip/hip_runtime.h>
#include <hip/hip_bf16.h>

#define BATCH 512
#define FEAT  512
#define GROUP 4

typedef __attribute__((ext_vector_type(2))) float v2f;
typedef __attribute__((ext_vector_type(8))) float v8f;

// -------- Kernel 1: sim[0:16, 0:512] = P[0:16,:] x P^T via V_WMMA_F32_16X16X4_F32
// One wave per 16-wide N tile; 32 waves total (4 blocks x 256 threads).
// Only rows 0..3 of the 16x16 D tiles are stored (that's all SmoothAP needs).
__global__ void __launch_bounds__(256) smoothap_gemm_wmma(const float* __restrict__ P,
                                                          float* __restrict__ sim) {
    const int lane = threadIdx.x & 31;
    const int wave = blockIdx.x * (blockDim.x >> 5) + (threadIdx.x >> 5);
    const int n0   = wave << 4;            // N-tile base (0..496)
    const int r    = lane & 15;            // row-in-tile selector
    const int hi   = lane >> 4;            // K half: 0 -> K=k0,k0+1 ; 1 -> K=k0+2,k0+3

    const float* __restrict__ arow = P + r * FEAT;          // A[m,k] = P[m,k]
    const float* __restrict__ brow = P + (n0 + r) * FEAT;   // B[k,n] = P[n,k]

    v8f c = {};
#pragma unroll 8
    for (int k0 = 0; k0 < FEAT; k0 += 4) {
        const int kc = k0 + (hi << 1);
        v2f a = *(const v2f*)(arow + kc);   // aligned float2 (kc even)
        v2f b = *(const v2f*)(brow + kc);
        // 8 args: (neg_a, A, neg_b, B, c_mod, C, reuse_a, reuse_b)
        c = __builtin_amdgcn_wmma_f32_16x16x4_f32(false, a, false, b,
                                                  (short)0, c, false, false);
    }

    // D layout: VGPR g, lanes 0-15 -> M=g, N=n0+lane. Need only M=0..3.
    if (lane < 16) {
        sim[0 * BATCH + n0 + lane] = c[0];
        sim[1 * BATCH + n0 + lane] = c[1];
        sim[2 * BATCH + n0 + lane] = c[2];
        sim[3 * BATCH + n0 + lane] = c[3];
    }
}

__device__ __forceinline__ float sigmoid_t100(float d) {
    // sigmoid_t(d, 0.01) = 1/(1+exp(clamp(-100*d, -50, 50)))
    float e = fminf(fmaxf(-d * 100.0f, -50.0f), 50.0f);
    return 1.0f / (1.0f + __expf(e));
}

// -------- Kernel 2: 16 waves, wave w handles pair (i=w/4, j=w%4):
// rk_all[i,j] = 1 + sum_{k!=j} sigmoid_t(sim[i,k]-sim[i,j]); then the 4x4
// positive part and the final scalar.
__global__ void __launch_bounds__(512) smoothap_reduce(const float* __restrict__ sim,
                                                       float* __restrict__ out) {
    __shared__ float rk_all[16];
    const int t    = threadIdx.x;
    const int lane = t & 31;
    const int w    = t >> 5;            // 0..15
    const int i    = w >> 2;
    const int j    = w & 3;

    const float sij = sim[i * BATCH + j];
    float acc = 0.0f;
#pragma unroll 4
    for (int k = lane; k < BATCH; k += 32) {
        if (k != j) acc += sigmoid_t100(sim[i * BATCH + k] - sij);
    }
    // wave32 butterfly reduction
#pragma unroll
    for (int off = 16; off > 0; off >>= 1)
        acc += __shfl_xor(acc, off, 32);
    if (lane == 0) rk_all[w] = acc + 1.0f;
    __syncthreads();

    if (t == 0) {
        float sp[GROUP][GROUP];
#pragma unroll
        for (int a = 0; a < GROUP; ++a)
#pragma unroll
            for (int b = 0; b < GROUP; ++b)
                sp[a][b] = sim[a * BATCH + b];   // sim_pos == sim_all[:4,:4]

        float pos_div = 0.0f;
#pragma unroll
        for (int a = 0; a < GROUP; ++a) {
#pragma unroll
            for (int b = 0; b < GROUP; ++b) {
                float rkp = 1.0f;
#pragma unroll
                for (int k = 0; k < GROUP; ++k) {
                    if (k == b) continue;
                    rkp += sigmoid_t100(sp[a][k] - sp[a][b]);
                }
                pos_div += rkp / rk_all[a * 4 + b];
            }
        }
        out[0] = 1.0f - pos_div / (float)(GROUP * GROUP);
    }
}

extern "C" void kernel_launch(void* const* d_in, const int* in_sizes, int n_in,
                              void* d_out, int out_size, void* d_ws, size_t ws_size,
                              hipStream_t stream) {
    (void)in_sizes; (void)n_in; (void)out_size; (void)ws_size;
    const float* preds = (const float*)d_in[0];
    float* out = (float*)d_out;
    float* sim = (float*)d_ws;   // 4 x 512 floats = 8 KB scratch

    // 32 waves cover N=512 in 16-wide tiles: 4 blocks x 256 threads (8 waves/block)
    smoothap_gemm_wmma<<<4, 256, 0, stream>>>(preds, sim);
    // single block, 16 wave32s
    smoothap_reduce<<<1, 512, 0, stream>>>(sim, out);
}